// MentionRankingModel_49091476193753
// MI455X (gfx1250) — compile-verified
//
#include <hip/hip_runtime.h>
#include <hip/hip_bf16.h>

typedef __attribute__((ext_vector_type(2))) float v2f;
typedef __attribute__((ext_vector_type(8))) float v8f;

#define N_MENTIONS 512
#define N_FEATS    20
#define HDIM       128
#define NCANDS     130816   // 512*511/2
#define TILE       16
#define STR        132      // LDS row stride in floats: 132 % 64 == 4 -> conflict-free WMMA reads

// ---------------------------------------------------------------------------
// Zero the 512x512 output (harness poisons it with 0xAA).
// ---------------------------------------------------------------------------
__global__ void zero_out_kernel(float* __restrict__ out, int n) {
    int i = blockIdx.x * blockDim.x + threadIdx.x;
    if (i < n) out[i] = 0.0f;
}

// ---------------------------------------------------------------------------
// Kernel A: h_a = tanh(sum_f emb_a[phi_a] + bias_a); write eps scores to the
// diagonal of out, and a_dot[i] = h_a[i] . w_ana[:128] + b_ana to workspace.
// One block (128 threads) per mention; thread t owns feature column t.
// ---------------------------------------------------------------------------
__global__ void __launch_bounds__(HDIM) ha_kernel(
    const int*   __restrict__ phi_a,
    const float* __restrict__ emb_a,
    const float* __restrict__ bias_a,
    const float* __restrict__ w_eps,
    const float* __restrict__ b_eps,
    const float* __restrict__ w_ana,
    const float* __restrict__ b_ana,
    float*       __restrict__ out,
    float*       __restrict__ a_dot)
{
    const int i = blockIdx.x;    // mention
    const int t = threadIdx.x;   // hidden column

    float sum = 0.0f;
#pragma unroll
    for (int f = 0; f < N_FEATS; ++f) {
        int idx = phi_a[i * N_FEATS + f];
        if (idx != 0)  // padding_idx = 0
            sum += emb_a[(long)idx * HDIM + t];
    }
    float h = tanhf(sum + bias_a[t]);

    __shared__ float se[HDIM];
    __shared__ float sa[HDIM];
    se[t] = h * w_eps[t];
    sa[t] = h * w_ana[t];
    __syncthreads();
#pragma unroll
    for (int s = HDIM / 2; s > 0; s >>= 1) {
        if (t < s) { se[t] += se[t + s]; sa[t] += sa[t + s]; }
        __syncthreads();
    }
    if (t == 0) {
        out[i * N_MENTIONS + i] = se[0] + b_eps[0];
        a_dot[i]                = sa[0] + b_ana[0];  // fold b_ana in here
    }
}

// ---------------------------------------------------------------------------
// Kernel B: the hot loop. One block (128 threads) handles 16 candidates.
// Phase 1: thread t accumulates column t of h_p for all 16 candidates
//          (gathers are 512B contiguous per index -> fully coalesced; the
//          102MB emb_p table is L2-resident on MI455X's 192MB L2).
// Phase 2: wave 0 computes the 16 dot products h_p . w_ana[128:] with 32
//          chained V_WMMA_F32_16X16X4_F32 ops (B = w chunk broadcast to all
//          16 columns), extracts per the 16x16 f32 C/D lane layout, and
//          scatters ana scores into the strict lower triangle of out.
// ---------------------------------------------------------------------------
__global__ void __launch_bounds__(128) cand_kernel(
    const int*   __restrict__ phi_p,
    const float* __restrict__ emb_p,
    const float* __restrict__ bias_p,
    const float* __restrict__ w_ana,
    const float* __restrict__ a_dot,
    float*       __restrict__ out)
{
    __shared__ float sH[TILE * STR];        // h_p tile, padded stride
    __shared__ int   sIdx[TILE * N_FEATS];  // candidate feature indices
    __shared__ float sW[HDIM];              // w_ana[128:]

    const int  t    = threadIdx.x;
    const long base = (long)blockIdx.x * TILE;

    // stage indices and the projection vector
    for (int k = t; k < TILE * N_FEATS; k += 128)
        sIdx[k] = phi_p[base * N_FEATS + k];
    if (t < HDIM) sW[t] = w_ana[HDIM + t];
    __syncthreads();

    const float bp = bias_p[t];
#pragma unroll 4
    for (int ci = 0; ci < TILE; ++ci) {
        float sum = 0.0f;
#pragma unroll
        for (int f = 0; f < N_FEATS; ++f) {
            int idx = sIdx[ci * N_FEATS + f];
            if (idx != 0)
                sum += emb_p[(long)idx * HDIM + t];
        }
        sH[ci * STR + t] = tanhf(sum + bp);
    }
    __syncthreads();

    // Wave 0 only: EXEC is all-1s for lanes 0..31 inside this branch.
    if (t < 32) {
        const int lane  = t;
        const int m     = lane & 15;               // A-matrix row owned by lane
        const int khalf = (lane & 16) ? 2 : 0;     // K offset per ISA 16x4 f32 layout

        v8f acc = {0.f, 0.f, 0.f, 0.f, 0.f, 0.f, 0.f, 0.f};
#pragma unroll
        for (int kc = 0; kc < 32; ++kc) {          // 128 = 32 * K4
            const int k0 = 4 * kc + khalf;
            v2f a, b;
            a.x = sH[m * STR + k0];
            a.y = sH[m * STR + k0 + 1];
            b.x = sW[k0];                          // broadcast w chunk to all N
            b.y = sW[k0 + 1];
            acc = __builtin_amdgcn_wmma_f32_16x16x4_f32(
                false, a, false, b, (short)0, acc, false, false);
        }

        // D[m][n] identical across n. Layout: VGPR r, lanes 0-15 -> M=r,
        // lanes 16-31 -> M=r+8. Lanes 0..7 and 16..23 each extract one row.
        int   mm  = -1;
        float dot = 0.0f;
        if (lane < 8)                       { mm = lane;     dot = acc[lane];      }
        else if (lane >= 16 && lane < 24)   { mm = lane - 8; dot = acc[lane - 16]; }

        if (mm >= 0) {
            const long c = base + mm;
            // anaphor j: largest j with j*(j-1)/2 <= c ; exact fixup after sqrt
            int j = (int)((1.0f + sqrtf(8.0f * (float)c + 1.0f)) * 0.5f);
            while ((long)j * (j - 1) / 2 > c)  --j;
            while ((long)(j + 1) * j / 2 <= c) ++j;
            const int col = (int)(c - (long)j * (j - 1) / 2);
            out[j * N_MENTIONS + col] = dot + a_dot[j];
        }
    }
}

// ---------------------------------------------------------------------------
extern "C" void kernel_launch(void* const* d_in, const int* in_sizes, int n_in,
                              void* d_out, int out_size, void* d_ws, size_t ws_size,
                              hipStream_t stream) {
    const int*   phi_a  = (const int*)  d_in[0];
    const int*   phi_p  = (const int*)  d_in[1];
    const float* emb_a  = (const float*)d_in[2];
    const float* bias_a = (const float*)d_in[3];
    const float* emb_p  = (const float*)d_in[4];
    const float* bias_p = (const float*)d_in[5];
    const float* w_eps  = (const float*)d_in[6];
    const float* b_eps  = (const float*)d_in[7];
    const float* w_ana  = (const float*)d_in[8];
    const float* b_ana  = (const float*)d_in[9];

    float* out   = (float*)d_out;
    float* a_dot = (float*)d_ws;   // 512 floats of scratch

    const int nout = N_MENTIONS * N_MENTIONS;
    zero_out_kernel<<<(nout + 255) / 256, 256, 0, stream>>>(out, nout);

    ha_kernel<<<N_MENTIONS, HDIM, 0, stream>>>(
        phi_a, emb_a, bias_a, w_eps, b_eps, w_ana, b_ana, out, a_dot);

    cand_kernel<<<NCANDS / TILE, 128, 0, stream>>>(
        phi_p, emb_p, bias_p, w_ana, a_dot, out);
}